// GraphEncoder_14998025798102
// MI455X (gfx1250) — compile-verified
//
#include <hip/hip_runtime.h>
#include <math.h>

#define N_NODES   50000
#define N_EDGES   1600000
#define N_GRAPHS  64
#define IN_DIM    64
#define HIDDEN    64
#define HEADS     4
#define HEAD_DIM  16
#define OUT_DIM   128
#define NEG_SLOPE 0.2f
#define E_TOTAL   (N_EDGES + N_NODES)   // self-loops appended

typedef __attribute__((ext_vector_type(2))) float v2f;
typedef __attribute__((ext_vector_type(8))) float v8f;

// -------------------- helpers --------------------

__device__ __forceinline__ void atomicMaxFloat(float* addr, float val) {
    // Monotonic int/uint punning trick; buffer must be initialized to -inf.
    if (val >= 0.0f) atomicMax((int*)addr, __float_as_int(val));
    else             atomicMin((unsigned int*)addr, __float_as_uint(val));
}

__global__ void k_fill(float* __restrict__ p, float v, int n) {
    int i = blockIdx.x * blockDim.x + threadIdx.x;
    if (i < n) p[i] = v;
}

// -------------------- WMMA dual linear: xl = x@Wl+bl, xr = x@Wr+br --------------------
// grid.x = N_NODES/16 (=3125), block = 128 threads = 4 waves; wave w owns cols [16w,16w+16)
__global__ void k_dual_linear(const float* __restrict__ x,
                              const float* __restrict__ Wl, const float* __restrict__ bl,
                              const float* __restrict__ Wr, const float* __restrict__ br,
                              float* __restrict__ xl, float* __restrict__ xr)
{
    int wave  = threadIdx.x >> 5;
    int lane  = threadIdx.x & 31;
    int row0  = blockIdx.x * 16;
    int n0    = wave * 16;
    int mrow  = lane & 15;          // A: row, B: col, C: col
    int khalf = (lane >> 4) * 2;    // K sub-offset per lane half

    v8f cl = {}; v8f cr = {};
    const float* arow = x + (size_t)(row0 + mrow) * HIDDEN;

    #pragma unroll
    for (int k0 = 0; k0 < HIDDEN; k0 += 4) {
        v2f a;  a.x  = arow[k0 + khalf];
                a.y  = arow[k0 + khalf + 1];
        v2f bL; bL.x = Wl[(size_t)(k0 + khalf)     * HIDDEN + n0 + mrow];
                bL.y = Wl[(size_t)(k0 + khalf + 1) * HIDDEN + n0 + mrow];
        v2f bR; bR.x = Wr[(size_t)(k0 + khalf)     * HIDDEN + n0 + mrow];
                bR.y = Wr[(size_t)(k0 + khalf + 1) * HIDDEN + n0 + mrow];
        cl = __builtin_amdgcn_wmma_f32_16x16x4_f32(false, a, false, bL, (short)0, cl, false, false);
        cr = __builtin_amdgcn_wmma_f32_16x16x4_f32(false, a, false, bR, (short)0, cr, false, false);
    }

    float biasL = bl[n0 + mrow];
    float biasR = br[n0 + mrow];
    int rbase = (lane >> 4) * 8;
    #pragma unroll
    for (int r = 0; r < 8; r++) {
        size_t o = (size_t)(row0 + rbase + r) * HIDDEN + n0 + mrow;
        xl[o] = cl[r] + biasL;
        xr[o] = cr[r] + biasR;
    }
}

// -------------------- edge pass 1: logits + segment max --------------------
// one thread per (edge, head)
__global__ void k_edge_logits(const float* __restrict__ xl, const float* __restrict__ xr,
                              const int* __restrict__ ei, const float* __restrict__ att,
                              float* __restrict__ logits, float* __restrict__ m)
{
    int t = blockIdx.x * blockDim.x + threadIdx.x;
    if (t >= E_TOTAL * HEADS) return;
    int e = t >> 2, h = t & 3;
    int src, dst;
    if (e < N_EDGES) { src = ei[e]; dst = ei[N_EDGES + e]; }
    else             { src = dst = e - N_EDGES; }

    const float* pl = xl + (size_t)src * HIDDEN + h * HEAD_DIM;
    const float* pr = xr + (size_t)dst * HIDDEN + h * HEAD_DIM;
    const float* pa = att + h * HEAD_DIM;
    float acc = 0.0f;
    #pragma unroll
    for (int c = 0; c < HEAD_DIM; c++) {
        float v = pl[c] + pr[c];
        v = v > 0.0f ? v : NEG_SLOPE * v;
        acc += pa[c] * v;
    }
    logits[t] = acc;
    atomicMaxFloat(&m[dst * HEADS + h], acc);
}

// -------------------- edge pass 2: exp + segment sum --------------------
__global__ void k_edge_exp(const int* __restrict__ ei,
                           float* __restrict__ logits,        // in: logit, out: ex
                           const float* __restrict__ m, float* __restrict__ denom)
{
    int t = blockIdx.x * blockDim.x + threadIdx.x;
    if (t >= E_TOTAL * HEADS) return;
    int e = t >> 2, h = t & 3;
    int dst = (e < N_EDGES) ? ei[N_EDGES + e] : (e - N_EDGES);
    float ex = __expf(logits[t] - m[dst * HEADS + h]);
    logits[t] = ex;
    atomicAdd(&denom[dst * HEADS + h], ex);
}

// -------------------- edge pass 3: alpha-weighted scatter-add --------------------
__global__ void k_edge_agg(const float* __restrict__ xl, const int* __restrict__ ei,
                           const float* __restrict__ ex, const float* __restrict__ denom,
                           float* __restrict__ agg)
{
    int t = blockIdx.x * blockDim.x + threadIdx.x;
    if (t >= E_TOTAL * HEADS) return;
    int e = t >> 2, h = t & 3;
    int src, dst;
    if (e < N_EDGES) { src = ei[e]; dst = ei[N_EDGES + e]; }
    else             { src = dst = e - N_EDGES; }

    const float* pl = xl + (size_t)src * HIDDEN + h * HEAD_DIM;
    __builtin_prefetch(pl, 0, 1);   // global_prefetch_b8
    float alpha = ex[t] / (denom[dst * HEADS + h] + 1e-16f);
    float* pd = agg + (size_t)dst * HIDDEN + h * HEAD_DIM;
    #pragma unroll
    for (int c = 0; c < HEAD_DIM; c++)
        atomicAdd(&pd[c], pl[c] * alpha);
}

// -------------------- relu(agg + bias) in place --------------------
__global__ void k_relu_bias(float* __restrict__ agg, const float* __restrict__ bias) {
    int i = blockIdx.x * blockDim.x + threadIdx.x;
    if (i >= N_NODES * HIDDEN) return;
    float v = agg[i] + bias[i & (HIDDEN - 1)];
    agg[i] = v > 0.0f ? v : 0.0f;
}

// -------------------- global mean pool --------------------
__global__ void k_count(const int* __restrict__ batch, float* __restrict__ counts) {
    int i = blockIdx.x * blockDim.x + threadIdx.x;
    if (i < N_NODES) atomicAdd(&counts[batch[i]], 1.0f);
}
__global__ void k_poolsum(const float* __restrict__ x, const int* __restrict__ batch,
                          float* __restrict__ pooled) {
    int t = blockIdx.x * blockDim.x + threadIdx.x;
    if (t >= N_NODES * HIDDEN) return;
    int nd = t >> 6, c = t & 63;
    atomicAdd(&pooled[batch[nd] * HIDDEN + c], x[t]);
}
__global__ void k_gmean(const float* __restrict__ pooled, const float* __restrict__ counts,
                        float* __restrict__ g) {
    int i = blockIdx.x * blockDim.x + threadIdx.x;
    if (i >= N_GRAPHS * HIDDEN) return;
    float cnt = counts[i >> 6];
    g[i] = pooled[i] / (cnt > 1.0f ? cnt : 1.0f);
}

// -------------------- WMMA GEMM for the MLP head --------------------
// out[rows x NCOLS] = act(A[rows x K] @ W[K x NCOLS] + bias); block = 16 rows, wave = 16 cols
template <int K, int NCOLS, bool RELU>
__global__ void k_gemm_wmma(const float* __restrict__ A, const float* __restrict__ W,
                            const float* __restrict__ bias, float* __restrict__ out)
{
    int wave  = threadIdx.x >> 5;
    int lane  = threadIdx.x & 31;
    int row0  = blockIdx.x * 16;
    int n0    = wave * 16;
    int mrow  = lane & 15;
    int khalf = (lane >> 4) * 2;

    v8f c = {};
    const float* arow = A + (size_t)(row0 + mrow) * K;
    #pragma unroll
    for (int k0 = 0; k0 < K; k0 += 4) {
        v2f a; a.x = arow[k0 + khalf];
               a.y = arow[k0 + khalf + 1];
        v2f b; b.x = W[(size_t)(k0 + khalf)     * NCOLS + n0 + mrow];
               b.y = W[(size_t)(k0 + khalf + 1) * NCOLS + n0 + mrow];
        c = __builtin_amdgcn_wmma_f32_16x16x4_f32(false, a, false, b, (short)0, c, false, false);
    }
    float bv = bias[n0 + mrow];
    int rbase = (lane >> 4) * 8;
    #pragma unroll
    for (int r = 0; r < 8; r++) {
        float v = c[r] + bv;
        if (RELU) v = v > 0.0f ? v : 0.0f;
        out[(size_t)(row0 + rbase + r) * NCOLS + n0 + mrow] = v;
    }
}

// -------------------- host orchestration --------------------

static void run_gat_layer(const float* x, const int* ei,
                          const float* Wl, const float* bl,
                          const float* Wr, const float* br,
                          const float* att, const float* bias,
                          float* xl, float* xr, float* logits,
                          float* m, float* denom, float* agg,
                          hipStream_t stream)
{
    const int TPB = 256;
    const int EH  = E_TOTAL * HEADS;
    k_dual_linear<<<N_NODES / 16, 128, 0, stream>>>(x, Wl, bl, Wr, br, xl, xr);
    k_fill<<<(N_NODES * HEADS + TPB - 1) / TPB, TPB, 0, stream>>>(m, -INFINITY, N_NODES * HEADS);
    k_fill<<<(N_NODES * HEADS + TPB - 1) / TPB, TPB, 0, stream>>>(denom, 0.0f, N_NODES * HEADS);
    k_fill<<<(N_NODES * HIDDEN + TPB - 1) / TPB, TPB, 0, stream>>>(agg, 0.0f, N_NODES * HIDDEN);
    k_edge_logits<<<(EH + TPB - 1) / TPB, TPB, 0, stream>>>(xl, xr, ei, att, logits, m);
    k_edge_exp   <<<(EH + TPB - 1) / TPB, TPB, 0, stream>>>(ei, logits, m, denom);
    k_edge_agg   <<<(EH + TPB - 1) / TPB, TPB, 0, stream>>>(xl, ei, logits, denom, agg);
    k_relu_bias  <<<(N_NODES * HIDDEN + TPB - 1) / TPB, TPB, 0, stream>>>(agg, bias);
}

extern "C" void kernel_launch(void* const* d_in, const int* in_sizes, int n_in,
                              void* d_out, int out_size, void* d_ws, size_t ws_size,
                              hipStream_t stream)
{
    const float* x     = (const float*)d_in[0];
    const int*   ei    = (const int*)  d_in[1];
    const int*   batch = (const int*)  d_in[2];
    const float* Wl0 = (const float*)d_in[3],  *bl0 = (const float*)d_in[4];
    const float* Wr0 = (const float*)d_in[5],  *br0 = (const float*)d_in[6];
    const float* att0  = (const float*)d_in[7], *bias0 = (const float*)d_in[8];
    const float* Wl1 = (const float*)d_in[9],  *bl1 = (const float*)d_in[10];
    const float* Wr1 = (const float*)d_in[11], *br1 = (const float*)d_in[12];
    const float* att1  = (const float*)d_in[13], *bias1 = (const float*)d_in[14];
    const float* W1 = (const float*)d_in[15], *b1 = (const float*)d_in[16];
    const float* W2 = (const float*)d_in[17], *b2 = (const float*)d_in[18];

    const size_t NF = (size_t)N_NODES * HIDDEN;
    float* ws     = (float*)d_ws;
    float* xl     = ws;
    float* xr     = xl   + NF;
    float* agg0   = xr   + NF;
    float* agg1   = agg0 + NF;
    float* logits = agg1 + NF;                       // E_TOTAL*HEADS
    float* m      = logits + (size_t)E_TOTAL * HEADS;
    float* denom  = m     + (size_t)N_NODES * HEADS;
    float* pooled = denom + (size_t)N_NODES * HEADS; // N_GRAPHS*HIDDEN
    float* counts = pooled + N_GRAPHS * HIDDEN;      // N_GRAPHS
    float* g      = counts + N_GRAPHS;               // N_GRAPHS*HIDDEN
    float* h1     = g + N_GRAPHS * HIDDEN;           // N_GRAPHS*OUT_DIM

    // layer 0 (input = node features), layer 1 (input = agg0)
    run_gat_layer(x,    ei, Wl0, bl0, Wr0, br0, att0, bias0, xl, xr, logits, m, denom, agg0, stream);
    run_gat_layer(agg0, ei, Wl1, bl1, Wr1, br1, att1, bias1, xl, xr, logits, m, denom, agg1, stream);

    // global mean pool
    const int TPB = 256;
    k_fill<<<(N_GRAPHS * HIDDEN + N_GRAPHS + TPB - 1) / TPB, TPB, 0, stream>>>(
        pooled, 0.0f, N_GRAPHS * HIDDEN + N_GRAPHS);
    k_count<<<(N_NODES + TPB - 1) / TPB, TPB, 0, stream>>>(batch, counts);
    k_poolsum<<<(N_NODES * HIDDEN + TPB - 1) / TPB, TPB, 0, stream>>>(agg1, batch, pooled);
    k_gmean<<<(N_GRAPHS * HIDDEN + TPB - 1) / TPB, TPB, 0, stream>>>(pooled, counts, g);

    // MLP head: h1 = relu(g@W1+b1), out = h1@W2+b2   (64 rows -> 4 row-tiles; 128 cols -> 8 waves)
    k_gemm_wmma<HIDDEN,  OUT_DIM, true ><<<N_GRAPHS / 16, 256, 0, stream>>>(g,  W1, b1, h1);
    k_gemm_wmma<OUT_DIM, OUT_DIM, false><<<N_GRAPHS / 16, 256, 0, stream>>>(h1, W2, b2, (float*)d_out);
}